// ContrastiveLoss_45466523796029
// MI455X (gfx1250) — compile-verified
//
#include <hip/hip_runtime.h>
#include <hip/hip_bf16.h>

// Problem constants (match reference: prediction (2,16,32,256,256) f32, gt (2,32,256,256) int)
#define N_VOX   4194304u      // 2 * 32 * 256 * 256 voxels
#define PLANE   2097152u      // 32*256*256 = per-batch voxel count = channel stride
#define CH_SH   21            // log2(PLANE)
#define NCH     16
#define NLAB    51

// Workspace layout (float offsets)
#define WS_SUMS    0          // 51*16 = 816
#define WS_COUNTS  816        // 51
#define WS_INTRA   867        // 51
#define WS_MEANS   918        // 816
#define WS_MNORM   1734       // 51
#define WS_CMN     1788       // 64*16 = 1024 (padded, 16B-aligned rows)
#define WS_TOTAL   2812

typedef __attribute__((ext_vector_type(2))) float v2f;
typedef __attribute__((ext_vector_type(8))) float v8f;

// ---------------- zero workspace (re-run every launch: deterministic) -------
__global__ void cl_zero_ws(float* __restrict__ ws) {
    for (unsigned t = blockIdx.x * blockDim.x + threadIdx.x; t < WS_TOTAL;
         t += gridDim.x * blockDim.x)
        ws[t] = 0.0f;
}

// ---------------- pass 1: per-label sums + counts ---------------------------
__global__ void cl_pass1(const float* __restrict__ pred,
                         const int* __restrict__ gt,
                         float* __restrict__ ws) {
    __shared__ float lacc[867];               // [0..815] sums, [816..866] counts
    for (int t = threadIdx.x; t < 867; t += blockDim.x) lacc[t] = 0.0f;
    __syncthreads();

    const unsigned stride = gridDim.x * blockDim.x;
    for (unsigned n = blockIdx.x * blockDim.x + threadIdx.x; n < N_VOX; n += stride) {
        int id = gt[n];
        id = id < 0 ? 0 : (id > NLAB - 1 ? NLAB - 1 : id);
        const unsigned b    = n >> CH_SH;
        const unsigned rem  = n & (PLANE - 1u);
        const unsigned base = (b << 25) + rem;          // b * 16 * PLANE + rem
        float* s = &lacc[id * NCH];
        #pragma unroll
        for (int ch = 0; ch < NCH; ++ch)
            atomicAdd(&s[ch], pred[base + ((unsigned)ch << CH_SH)]);  // ds_add_f32
        atomicAdd(&lacc[816 + id], 1.0f);
    }
    __syncthreads();
    for (int t = threadIdx.x; t < 867; t += blockDim.x)
        unsafeAtomicAdd(&ws[WS_SUMS + t], lacc[t]);     // global_atomic_add_f32
}

// ---------------- tiny: means, norms, normalized padded cmn -----------------
__global__ void cl_stats(float* __restrict__ ws) {
    const int t = threadIdx.x;
    if (t >= NLAB) return;
    const float cnt = ws[WS_COUNTS + t];
    const float inv = 1.0f / fmaxf(cnt, 1.0f);
    float m[NCH];
    float n2 = 0.0f;
    #pragma unroll
    for (int ch = 0; ch < NCH; ++ch) {
        const float mv = ws[WS_SUMS + t * NCH + ch] * inv;
        m[ch] = mv;
        n2 = fmaf(mv, mv, n2);
        ws[WS_MEANS + t * NCH + ch] = mv;
    }
    const float nrm = sqrtf(n2);
    ws[WS_MNORM + t] = nrm;
    if (t >= 1) {                                        // cmn rows 0..49 (label 1..50)
        const float s = 1.0f / fmaxf(nrm, 1e-8f);
        #pragma unroll
        for (int ch = 0; ch < NCH; ++ch)
            ws[WS_CMN + (t - 1) * NCH + ch] = m[ch] * s;
    }
}

// ---------------- pass 2: per-voxel cosine, per-label sum -------------------
__global__ void cl_pass2(const float* __restrict__ pred,
                         const int* __restrict__ gt,
                         float* __restrict__ ws) {
    __shared__ float lmean[816];
    __shared__ float lnorm[NLAB];
    __shared__ float lcos[NLAB];
    for (int t = threadIdx.x; t < 816; t += blockDim.x) lmean[t] = ws[WS_MEANS + t];
    for (int t = threadIdx.x; t < NLAB; t += blockDim.x) {
        lnorm[t] = ws[WS_MNORM + t];
        lcos[t]  = 0.0f;
    }
    __syncthreads();

    const unsigned stride = gridDim.x * blockDim.x;
    for (unsigned n = blockIdx.x * blockDim.x + threadIdx.x; n < N_VOX; n += stride) {
        int id = gt[n];
        id = id < 0 ? 0 : (id > NLAB - 1 ? NLAB - 1 : id);
        const unsigned b    = n >> CH_SH;
        const unsigned rem  = n & (PLANE - 1u);
        const unsigned base = (b << 25) + rem;
        const float* mrow = &lmean[id * NCH];
        float dot = 0.0f, p2 = 0.0f;
        #pragma unroll
        for (int ch = 0; ch < NCH; ++ch) {
            const float v = pred[base + ((unsigned)ch << CH_SH)];
            dot = fmaf(v, mrow[ch], dot);
            p2  = fmaf(v, v, p2);
        }
        const float cosv = dot / fmaxf(sqrtf(p2) * lnorm[id], 1e-8f);
        atomicAdd(&lcos[id], cosv);
    }
    __syncthreads();
    for (int t = threadIdx.x; t < NLAB; t += blockDim.x)
        unsafeAtomicAdd(&ws[WS_INTRA + t], lcos[t]);
}

// ---------------- finalize: WMMA Gram (64x64 padded) + scalar loss ----------
__device__ __forceinline__ v2f mk2(float a, float b) { v2f r; r.x = a; r.y = b; return r; }

__global__ void cl_finalize(const float* __restrict__ ws, float* __restrict__ out) {
    const int  lane = threadIdx.x;        // one wave32, EXEC all ones
    const int  rlow = lane & 15;
    const bool hi   = lane >= 16;

    // A operand, 16x4 f32 layout: lanes 0-15 hold (K,K+1), lanes 16-31 hold (K+2,K+3)
    // of row M = lane&15. B (4x16) mirrors it with N on the lanes, so for G = M*M^T
    // the per-lane B data for row-block c equals the A data for row-block c.
    v2f op[4][4];
    #pragma unroll
    for (int r = 0; r < 4; ++r) {
        const float4* row = (const float4*)&ws[WS_CMN + (16 * r + rlow) * NCH];
        const float4 q0 = row[0], q1 = row[1], q2 = row[2], q3 = row[3];
        op[r][0] = hi ? mk2(q0.z, q0.w) : mk2(q0.x, q0.y);
        op[r][1] = hi ? mk2(q1.z, q1.w) : mk2(q1.x, q1.y);
        op[r][2] = hi ? mk2(q2.z, q2.w) : mk2(q2.x, q2.y);
        op[r][3] = hi ? mk2(q3.z, q3.w) : mk2(q3.x, q3.y);
    }

    float sum = 0.0f;   // sum of clip(G) over ALL off-diagonal entries (= 2x upper tri)
    #pragma unroll
    for (int ti = 0; ti < 4; ++ti) {
        #pragma unroll
        for (int tj = 0; tj < 4; ++tj) {
            v8f acc = {};
            acc = __builtin_amdgcn_wmma_f32_16x16x4_f32(false, op[ti][0], false, op[tj][0],
                                                        (short)0, acc, false, false);
            acc = __builtin_amdgcn_wmma_f32_16x16x4_f32(false, op[ti][1], false, op[tj][1],
                                                        (short)0, acc, false, false);
            acc = __builtin_amdgcn_wmma_f32_16x16x4_f32(false, op[ti][2], false, op[tj][2],
                                                        (short)0, acc, false, false);
            acc = __builtin_amdgcn_wmma_f32_16x16x4_f32(false, op[ti][3], false, op[tj][3],
                                                        (short)0, acc, false, false);
            // C/D layout: VGPR v, lanes 0-15 -> M=v, lanes 16-31 -> M=v+8, N = lane&15
            #pragma unroll
            for (int v = 0; v < 8; ++v) {
                const int rg = 16 * ti + v + (hi ? 8 : 0);
                const int cg = 16 * tj + rlow;
                float g = acc[v];
                g = fminf(fmaxf(g, 0.0f), 1.0f);
                sum += (rg != cg) ? g : 0.0f;
            }
        }
    }

    // wave32 reduction
    #pragma unroll
    for (int off = 16; off > 0; off >>= 1)
        sum += __shfl_xor(sum, off, 32);

    if (lane == 0) {
        float intra = 0.0f;
        for (int l = 1; l < NLAB; ++l)
            intra += ws[WS_INTRA + l] / fmaxf(ws[WS_COUNTS + l], 1.0f);
        intra *= (1.0f / 50.0f);
        const float inter = sum * (0.5f / 1225.0f);   // half (both triangles) / C(50,2)
        out[0] = inter - intra;
    }
}

// ---------------------------------------------------------------------------
extern "C" void kernel_launch(void* const* d_in, const int* in_sizes, int n_in,
                              void* d_out, int out_size, void* d_ws, size_t ws_size,
                              hipStream_t stream) {
    const float* pred = (const float*)d_in[0];
    const int*   gt   = (const int*)d_in[1];
    float*       out  = (float*)d_out;
    float*       ws   = (float*)d_ws;

    cl_zero_ws<<<4, 256, 0, stream>>>(ws);
    cl_pass1<<<2048, 256, 0, stream>>>(pred, gt, ws);
    cl_stats<<<1, 64, 0, stream>>>(ws);
    cl_pass2<<<2048, 256, 0, stream>>>(pred, gt, ws);
    cl_finalize<<<1, 32, 0, stream>>>(ws, out);
}